// DrugResponseModel_6588479832510
// MI455X (gfx1250) — compile-verified
//
#include <hip/hip_runtime.h>
#include <hip/hip_bf16.h>

// ---------------- model constants ----------------
#define B_      32
#define NGENES  20000
#define P_      300
#define G_      512
#define DIN_    10
#define KBIG    (G_ * DIN_)      // 5120
#define MROWS   (B_ * P_)        // 9600
#define GOUT    64
#define DOUT    64
#define ATTN_   256
#define HEADS_  4
#define DH_     64
#define HID_    1024
#define FINAL_  512
#define DRUGIN  768

typedef float v2f __attribute__((ext_vector_type(2)));
typedef float v8f __attribute__((ext_vector_type(8)));

// =====================================================================
// Gather: flat[b, p, g*10+d] = gene[b, max(idx[p,g],0), d]
// =====================================================================
__global__ void gather_kernel(const float* __restrict__ gene,
                              const int* __restrict__ idx,
                              float* __restrict__ flat)
{
    long t = (long)blockIdx.x * blockDim.x + threadIdx.x;
    const long total = (long)B_ * P_ * G_;
    if (t >= total) return;
    int g  = (int)(t % G_);
    long bp = t / G_;
    int p  = (int)(bp % P_);
    int b  = (int)(bp / P_);
    int gi = idx[p * G_ + g];
    if (gi < 0) gi = 0;
    const float* src = gene + ((size_t)b * NGENES + gi) * DIN_;
    float*       dst = flat + (size_t)bp * KBIG + (size_t)g * DIN_;
    // 40-byte chunks, 8-byte aligned on both sides -> 5 x b64
    const v2f* s2 = reinterpret_cast<const v2f*>(src);
    v2f*       d2 = reinterpret_cast<v2f*>(dst);
#pragma unroll
    for (int i = 0; i < 5; ++i) d2[i] = s2[i];
}

// =====================================================================
// FP32 WMMA GEMM: C[M x N] = A[M x K] * W[K x N] + bias
// One wave (32 threads) computes a 16 x (16*NT) tile using
// V_WMMA_F32_16X16X4_F32 with f32 accumulation (reference-exact dtype).
// Requires M%16==0, N%(16*NT)==0, K%4==0, lda/ldb/ldc even.
// =====================================================================
template <int NT>
__global__ __launch_bounds__(32)
void wmma_gemm_bias(const float* __restrict__ A, int lda,
                    const float* __restrict__ Bm, int ldb,
                    const float* __restrict__ bias,
                    float* __restrict__ C, int ldc, int K)
{
    const int m0   = blockIdx.x * 16;
    const int n0   = blockIdx.y * (16 * NT);
    const int lane = threadIdx.x;        // 0..31
    const int half = lane >> 4;          // 0: K{0,1}, 1: K{2,3}
    const int l15  = lane & 15;

    v8f zero = {};
    v8f acc[NT];
#pragma unroll
    for (int j = 0; j < NT; ++j) acc[j] = zero;

    // A fragment base: row m0+l15, K offset half*2 (VGPR0/1 = consecutive K)
    const float* aRow = A + (size_t)(m0 + l15) * lda + (half << 1);

    for (int k = 0; k < K; k += 4) {
        v2f a = *reinterpret_cast<const v2f*>(aRow + k);
        // B fragment: VGPR0 = row (k + 2*half), VGPR1 = next row, col n0+j*16+l15
        const float* bRow0 = Bm + (size_t)(k + (half << 1)) * ldb + n0 + l15;
        const float* bRow1 = bRow0 + ldb;
#pragma unroll
        for (int j = 0; j < NT; ++j) {
            v2f bf;
            bf.x = bRow0[j * 16];
            bf.y = bRow1[j * 16];
            acc[j] = __builtin_amdgcn_wmma_f32_16x16x4_f32(
                false, a, false, bf, (short)0, acc[j], false, false);
        }
    }

    // D layout: VGPR r -> rows m0 + half*8 + r, cols n0 + j*16 + l15
#pragma unroll
    for (int j = 0; j < NT; ++j) {
        int n    = n0 + j * 16 + l15;
        float bv = bias ? bias[n] : 0.0f;
#pragma unroll
        for (int r = 0; r < 8; ++r) {
            C[(size_t)(m0 + half * 8 + r) * ldc + n] = acc[j][r] + bv;
        }
    }
}

// =====================================================================
// Row LayerNorm (+optional ReLU). One block per row; blockDim = pow2.
// =====================================================================
__global__ void ln_kernel(const float* __restrict__ x, float* __restrict__ y,
                          const float* __restrict__ g, const float* __restrict__ b,
                          int cols, int relu)
{
    __shared__ float s1[256];
    __shared__ float s2[256];
    const int row = blockIdx.x;
    const int tid = threadIdx.x;
    const int bd  = blockDim.x;
    const float* xr = x + (size_t)row * cols;
    float sum = 0.0f, sq = 0.0f;
    for (int c = tid; c < cols; c += bd) {
        float v = xr[c];
        sum += v; sq += v * v;
    }
    s1[tid] = sum; s2[tid] = sq;
    __syncthreads();
    for (int s = bd >> 1; s > 0; s >>= 1) {
        if (tid < s) { s1[tid] += s1[tid + s]; s2[tid] += s2[tid + s]; }
        __syncthreads();
    }
    float mean = s1[0] / cols;
    float var  = s2[0] / cols - mean * mean;
    float r    = rsqrtf(var + 1e-5f);
    float* yr  = y + (size_t)row * cols;
    for (int c = tid; c < cols; c += bd) {
        float v = (xr[c] - mean) * r * g[c] + b[c];
        if (relu) v = fmaxf(v, 0.0f);
        yr[c] = v;
    }
}

// =====================================================================
// y[b,p,n] = qp[b,p,n] + o[b,n]   (residual + broadcast over p)
// =====================================================================
__global__ void bcast_add_kernel(const float* __restrict__ qp,
                                 const float* __restrict__ o,
                                 float* __restrict__ y, int Prows)
{
    size_t t = (size_t)blockIdx.x * blockDim.x + threadIdx.x;
    size_t total = (size_t)B_ * Prows * ATTN_;
    if (t >= total) return;
    int    n  = (int)(t & (ATTN_ - 1));
    size_t bp = t >> 8;                     // ATTN_==256
    int    b  = (int)(bp / Prows);
    y[t] = qp[t] + o[(size_t)b * ATTN_ + n];
}

// =====================================================================
// d2p attention (q length 1): one block per (b, head)
// =====================================================================
__global__ void d2p_attn_kernel(const float* __restrict__ q,    // (32,256)
                                const float* __restrict__ Km,   // (32,300,256)
                                const float* __restrict__ Vm,   // (32,300,256)
                                float* __restrict__ ctx,        // (32,256)
                                float* __restrict__ wOut)       // (32,4,300) or null
{
    __shared__ float s_q[DH_];
    __shared__ float s_sc[P_];
    __shared__ float s_red[128];
    __shared__ float s_part[2][DH_];
    const int b   = blockIdx.x >> 2;
    const int hh  = blockIdx.x & 3;
    const int tid = threadIdx.x;           // 128 threads

    if (tid < DH_) s_q[tid] = q[(size_t)b * ATTN_ + hh * DH_ + tid];
    __syncthreads();

    float lmax = -1e30f;
    for (int p = tid; p < P_; p += 128) {
        const float* krow = Km + ((size_t)b * P_ + p) * ATTN_ + hh * DH_;
        float dot = 0.0f;
#pragma unroll 8
        for (int d = 0; d < DH_; ++d) dot += s_q[d] * krow[d];
        dot *= 0.125f;                     // 1/sqrt(64)
        s_sc[p] = dot;
        lmax = fmaxf(lmax, dot);
    }
    s_red[tid] = lmax;
    __syncthreads();
    for (int s = 64; s > 0; s >>= 1) {
        if (tid < s) s_red[tid] = fmaxf(s_red[tid], s_red[tid + s]);
        __syncthreads();
    }
    float mx = s_red[0];
    __syncthreads();

    float lsum = 0.0f;
    for (int p = tid; p < P_; p += 128) {
        float e = expf(s_sc[p] - mx);
        s_sc[p] = e;
        lsum += e;
    }
    s_red[tid] = lsum;
    __syncthreads();
    for (int s = 64; s > 0; s >>= 1) {
        if (tid < s) s_red[tid] += s_red[tid + s];
        __syncthreads();
    }
    float inv = 1.0f / s_red[0];
    __syncthreads();

    for (int p = tid; p < P_; p += 128) {
        float w = s_sc[p] * inv;
        s_sc[p] = w;
        if (wOut) wOut[((size_t)b * HEADS_ + hh) * P_ + p] = w;
    }
    __syncthreads();

    // ctx: 128 threads = 2 key-chunks x 64 dims
    int d = tid & 63, chunk = tid >> 6;
    float accv = 0.0f;
    for (int p = chunk * 150; p < (chunk + 1) * 150; ++p)
        accv += s_sc[p] * Vm[((size_t)b * P_ + p) * ATTN_ + hh * DH_ + d];
    s_part[chunk][d] = accv;
    __syncthreads();
    if (tid < DH_)
        ctx[(size_t)b * ATTN_ + hh * DH_ + tid] = s_part[0][tid] + s_part[1][tid];
}

// =====================================================================
// comb[b, 0:256] = max_p q2p[b,p,:], comb[b,256:512] = q2d[b,:]
// =====================================================================
__global__ void maxconcat_kernel(const float* __restrict__ q2p,
                                 const float* __restrict__ q2d,
                                 float* __restrict__ comb)
{
    int t = blockIdx.x * blockDim.x + threadIdx.x;
    if (t >= B_ * 2 * ATTN_) return;
    int b = t >> 9;
    int n = t & 511;
    if (n < ATTN_) {
        float m = -1e30f;
        const float* base = q2p + (size_t)b * P_ * ATTN_ + n;
        for (int p = 0; p < P_; ++p) m = fmaxf(m, base[(size_t)p * ATTN_]);
        comb[t] = m;
    } else {
        comb[t] = q2d[(size_t)b * ATTN_ + (n - ATTN_)];
    }
}

// =====================================================================
// BN-eval + ReLU: x = relu(x * g*rsqrt(1+eps) + b)
// =====================================================================
__global__ void bnrelu_kernel(float* __restrict__ x, const float* __restrict__ g,
                              const float* __restrict__ b, int n, int total)
{
    int t = blockIdx.x * blockDim.x + threadIdx.x;
    if (t >= total) return;
    int c = t % n;
    float s = g[c] * rsqrtf(1.0f + 1e-5f);
    x[t] = fmaxf(0.0f, x[t] * s + b[c]);
}

__global__ void fill_kernel(float* __restrict__ p, int total, float v)
{
    int t = blockIdx.x * blockDim.x + threadIdx.x;
    if (t < total) p[t] = v;
}

// fc3: out[b] = dot(x2[b,:256], W) + bias
__global__ void fc3_kernel(const float* __restrict__ x2, const float* __restrict__ W,
                           const float* __restrict__ b3, float* __restrict__ out)
{
    int b = threadIdx.x;
    if (b < B_) {
        float s = 0.0f;
        for (int c = 0; c < FINAL_ / 2; ++c) s += x2[(size_t)b * (FINAL_ / 2) + c] * W[c];
        out[b] = s + b3[0];
    }
}

// =====================================================================
extern "C" void kernel_launch(void* const* d_in, const int* in_sizes, int n_in,
                              void* d_out, int out_size, void* d_ws, size_t ws_size,
                              hipStream_t stream)
{
    (void)in_sizes; (void)n_in; (void)out_size; (void)ws_size;
    auto F = [&](int k) { return (const float*)d_in[k]; };

    // ---- inputs (setup_inputs insertion order; params flattened in source order)
    const float* gene = F(0);                 // (32,20000,10)
    const float* drug = F(1);                 // (32,768)
    const int*   idx  = (const int*)d_in[2];  // (300,512)
    // gene_ffn
    const float *gW1 = F(3),  *gb1 = F(4),  *glng = F(5), *glnb = F(6);
    const float *gW2 = F(7),  *gb2 = F(8);
    const float *gvg = F(9),  *gvb = F(10);
    // drug_ffn
    const float *dW1 = F(11), *db1 = F(12), *dlng = F(13), *dlnb = F(14);
    const float *dW2 = F(15), *db2 = F(16);
    // p2d layers: q,k,v,o each W,b   (k-proj unused: softmax over 1 key == 1)
    const float *p0qW=F(17),*p0qb=F(18),             *p0vW=F(21),*p0vb=F(22),*p0oW=F(23),*p0ob=F(24);
    const float *p1qW=F(25),*p1qb=F(26),             *p1vW=F(29),*p1vb=F(30),*p1oW=F(31),*p1ob=F(32);
    // d2p layers
    const float *d0qW=F(33),*d0qb=F(34),*d0kW=F(35),*d0kb=F(36),*d0vW=F(37),*d0vb=F(38),*d0oW=F(39),*d0ob=F(40);
    const float *d1qW=F(41),*d1qb=F(42),*d1kW=F(43),*d1kb=F(44),*d1vW=F(45),*d1vb=F(46),*d1oW=F(47),*d1ob=F(48);
    // head
    const float *fc1W=F(49),*fc1b=F(50),*bn1g=F(51),*bn1b=F(52);
    const float *fc2W=F(53),*fc2b=F(54),*bn2g=F(55),*bn2b=F(56);
    const float *fc3W=F(57),*fc3b=F(58);

    // ---- outputs (concatenated flat, return order)
    float* out    = (float*)d_out;
    float* o_x    = out;               // 32
    float* o_p2dw = out + 32;          // 38400 (all ones)
    float* o_d2pw = out + 38432;       // 38400
    float* o_gev  = out + 76832;       // (32,300,64)
    float* o_drug = out + 691232;      // (32,64)
    float* o_pe   = out + 693280;      // (32,300,64)

    // ---- workspace layout (floats)
    float* ws     = (float*)d_ws;
    float* flatA  = ws;                          // 9600 x 5120
    float* h      = flatA + (size_t)MROWS * KBIG;          // 9600 x 1024
    float* qp_p   = h     + (size_t)MROWS * HID_;          // 9600 x 256
    float* qcur_p = qp_p  + (size_t)MROWS * ATTN_;         // 9600 x 256
    float* Kbuf   = qcur_p+ (size_t)MROWS * ATTN_;         // 9600 x 256
    float* Vbuf   = Kbuf  + (size_t)MROWS * ATTN_;         // 9600 x 256
    float* hd     = Vbuf  + (size_t)MROWS * ATTN_;         // 32 x 1024
    float* qp_d   = hd    + (size_t)B_ * HID_;             // 32 x 256
    float* qcur_d = qp_d  + (size_t)B_ * ATTN_;
    float* v_d    = qcur_d+ (size_t)B_ * ATTN_;
    float* o_d    = v_d   + (size_t)B_ * ATTN_;
    float* ctx_d  = o_d   + (size_t)B_ * ATTN_;
    float* comb   = ctx_d + (size_t)B_ * ATTN_;            // 32 x 512
    float* x1     = comb  + (size_t)B_ * 2 * ATTN_;        // 32 x 512
    float* x2     = x1    + (size_t)B_ * FINAL_;           // 32 x 256

    // ================= gene pathway FFN =================
    {
        long total = (long)B_ * P_ * G_;
        gather_kernel<<<(unsigned)((total + 255) / 256), 256, 0, stream>>>(gene, idx, flatA);
    }
    wmma_gemm_bias<8><<<dim3(MROWS / 16, HID_ / 128), 32, 0, stream>>>(
        flatA, KBIG, gW1, HID_, gb1, h, HID_, KBIG);
    ln_kernel<<<MROWS, 256, 0, stream>>>(h, h, glng, glnb, HID_, 1);
    wmma_gemm_bias<4><<<dim3(MROWS / 16, 1), 32, 0, stream>>>(
        h, HID_, gW2, GOUT, gb2, o_gev, GOUT, HID_);
    ln_kernel<<<MROWS, 64, 0, stream>>>(o_gev, o_pe, gvg, gvb, GOUT, 0);

    // ================= drug FFN =================
    wmma_gemm_bias<4><<<dim3(B_ / 16, HID_ / 64), 32, 0, stream>>>(
        drug, DRUGIN, dW1, HID_, db1, hd, HID_, DRUGIN);
    ln_kernel<<<B_, 256, 0, stream>>>(hd, hd, dlng, dlnb, HID_, 1);
    wmma_gemm_bias<4><<<dim3(B_ / 16, 1), 32, 0, stream>>>(
        hd, HID_, dW2, DOUT, db2, o_drug, DOUT, HID_);

    // ================= p2d cross-attn (kv len == 1 -> weights == 1) ====
    // layer 0
    wmma_gemm_bias<4><<<dim3(MROWS / 16, ATTN_ / 64), 32, 0, stream>>>(
        o_pe, GOUT, p0qW, ATTN_, p0qb, qp_p, ATTN_, GOUT);
    wmma_gemm_bias<4><<<dim3(B_ / 16, ATTN_ / 64), 32, 0, stream>>>(
        o_drug, DOUT, p0vW, ATTN_, p0vb, v_d, ATTN_, DOUT);
    wmma_gemm_bias<4><<<dim3(B_ / 16, ATTN_ / 64), 32, 0, stream>>>(
        v_d, ATTN_, p0oW, ATTN_, p0ob, o_d, ATTN_, ATTN_);
    bcast_add_kernel<<<(MROWS * ATTN_) / 256, 256, 0, stream>>>(qp_p, o_d, qcur_p, P_);
    // layer 1
    wmma_gemm_bias<4><<<dim3(MROWS / 16, ATTN_ / 64), 32, 0, stream>>>(
        qcur_p, ATTN_, p1qW, ATTN_, p1qb, qp_p, ATTN_, ATTN_);
    wmma_gemm_bias<4><<<dim3(B_ / 16, ATTN_ / 64), 32, 0, stream>>>(
        o_drug, DOUT, p1vW, ATTN_, p1vb, v_d, ATTN_, DOUT);
    wmma_gemm_bias<4><<<dim3(B_ / 16, ATTN_ / 64), 32, 0, stream>>>(
        v_d, ATTN_, p1oW, ATTN_, p1ob, o_d, ATTN_, ATTN_);
    bcast_add_kernel<<<(MROWS * ATTN_) / 256, 256, 0, stream>>>(qp_p, o_d, qcur_p, P_);
    fill_kernel<<<(B_ * HEADS_ * P_) / 256, 256, 0, stream>>>(o_p2dw, B_ * HEADS_ * P_, 1.0f);

    // ================= d2p cross-attn =================
    // layer 0
    wmma_gemm_bias<4><<<dim3(B_ / 16, ATTN_ / 64), 32, 0, stream>>>(
        o_drug, DOUT, d0qW, ATTN_, d0qb, qp_d, ATTN_, DOUT);
    wmma_gemm_bias<4><<<dim3(MROWS / 16, ATTN_ / 64), 32, 0, stream>>>(
        o_pe, GOUT, d0kW, ATTN_, d0kb, Kbuf, ATTN_, GOUT);
    wmma_gemm_bias<4><<<dim3(MROWS / 16, ATTN_ / 64), 32, 0, stream>>>(
        o_pe, GOUT, d0vW, ATTN_, d0vb, Vbuf, ATTN_, GOUT);
    d2p_attn_kernel<<<B_ * HEADS_, 128, 0, stream>>>(qp_d, Kbuf, Vbuf, ctx_d, nullptr);
    wmma_gemm_bias<4><<<dim3(B_ / 16, ATTN_ / 64), 32, 0, stream>>>(
        ctx_d, ATTN_, d0oW, ATTN_, d0ob, o_d, ATTN_, ATTN_);
    bcast_add_kernel<<<(B_ * ATTN_) / 256, 256, 0, stream>>>(qp_d, o_d, qcur_d, 1);
    // layer 1
    wmma_gemm_bias<4><<<dim3(B_ / 16, ATTN_ / 64), 32, 0, stream>>>(
        qcur_d, ATTN_, d1qW, ATTN_, d1qb, qp_d, ATTN_, ATTN_);
    wmma_gemm_bias<4><<<dim3(MROWS / 16, ATTN_ / 64), 32, 0, stream>>>(
        o_pe, GOUT, d1kW, ATTN_, d1kb, Kbuf, ATTN_, GOUT);
    wmma_gemm_bias<4><<<dim3(MROWS / 16, ATTN_ / 64), 32, 0, stream>>>(
        o_pe, GOUT, d1vW, ATTN_, d1vb, Vbuf, ATTN_, GOUT);
    d2p_attn_kernel<<<B_ * HEADS_, 128, 0, stream>>>(qp_d, Kbuf, Vbuf, ctx_d, o_d2pw);
    wmma_gemm_bias<4><<<dim3(B_ / 16, ATTN_ / 64), 32, 0, stream>>>(
        ctx_d, ATTN_, d1oW, ATTN_, d1ob, o_d, ATTN_, ATTN_);
    bcast_add_kernel<<<(B_ * ATTN_) / 256, 256, 0, stream>>>(qp_d, o_d, qcur_d, 1);

    // ================= head =================
    maxconcat_kernel<<<(B_ * 2 * ATTN_) / 256, 256, 0, stream>>>(qcur_p, qcur_d, comb);
    wmma_gemm_bias<4><<<dim3(B_ / 16, FINAL_ / 64), 32, 0, stream>>>(
        comb, 2 * ATTN_, fc1W, FINAL_, fc1b, x1, FINAL_, 2 * ATTN_);
    bnrelu_kernel<<<(B_ * FINAL_) / 256, 256, 0, stream>>>(x1, bn1g, bn1b, FINAL_, B_ * FINAL_);
    wmma_gemm_bias<4><<<dim3(B_ / 16, (FINAL_ / 2) / 64), 32, 0, stream>>>(
        x1, FINAL_, fc2W, FINAL_ / 2, fc2b, x2, FINAL_ / 2, FINAL_);
    bnrelu_kernel<<<(B_ * FINAL_ / 2) / 256, 256, 0, stream>>>(x2, bn2g, bn2b, FINAL_ / 2, B_ * FINAL_ / 2);
    fc3_kernel<<<1, 32, 0, stream>>>(x2, fc3W, fc3b, o_x);
}